// SimpleEncoder_55757265436720
// MI455X (gfx1250) — compile-verified
//
#include <hip/hip_runtime.h>

// CDNA5 (gfx1250) fused LSTM + output projection.
// One wave32 owns 16 batch rows for all 512 timesteps.
// Recurrent + input + output matmuls all via V_WMMA_F32_16X16X4_F32 (exact fp32).
// v3: hardware v_tanh_f32 nonlinearities (guarded by __has_builtin),
//     prefetch folded onto the pipelined load pointer.

typedef float v2f __attribute__((ext_vector_type(2)));
typedef float v8f __attribute__((ext_vector_type(8)));

#define WMMA_F32_16x16x4(a, b, c) \
  __builtin_amdgcn_wmma_f32_16x16x4_f32(false, (a), false, (b), (short)0, (c), false, false)

constexpr int I_ = 3;     // input features
constexpr int H_ = 16;    // hidden size
constexpr int O_ = 5;     // output size
constexpr int S_ = 512;   // sequence length

#if __has_builtin(__builtin_amdgcn_tanhf)
// gfx1250 has V_TANH_F32: 1 transcendental op per tanh,
// sigmoid(x) = 0.5*tanh(x/2)+0.5 -> mul + tanh + fma.
__device__ __forceinline__ float tanh_f(float v) {
  return __builtin_amdgcn_tanhf(v);
}
__device__ __forceinline__ float sigmoid_f(float v) {
  return __builtin_fmaf(0.5f, __builtin_amdgcn_tanhf(0.5f * v), 0.5f);
}
#else
__device__ __forceinline__ float fast_rcp(float v) {
  return __builtin_amdgcn_rcpf(v);          // v_rcp_f32
}
__device__ __forceinline__ float sigmoid_f(float v) {
  return fast_rcp(1.0f + __expf(-v));
}
__device__ __forceinline__ float tanh_f(float v) {
  // 1 - 2/(e^{2v}+1): saturates correctly to +/-1, no inf/inf NaN
  float e = __expf(2.0f * v);
  return 1.0f - 2.0f * fast_rcp(e + 1.0f);
}
#endif

__global__ __launch_bounds__(128) void lstm_wmma_f32_kernel(
    const float* __restrict__ x,      // [B, S, I]
    const float* __restrict__ W_ih,   // [4H, I]
    const float* __restrict__ W_hh,   // [4H, H]
    const float* __restrict__ b_ih,   // [4H]
    const float* __restrict__ b_hh,   // [4H]
    const float* __restrict__ W_lin,  // [O, H]
    const float* __restrict__ b_lin,  // [O]
    float* __restrict__ out)          // [B, S, O]
{
  // Per-wave LDS scratch for the 16x16 h relayout (C-layout -> A-layout).
  __shared__ float lds_h[4][16 * 16];

  const int lane        = threadIdx.x & 31;
  const int wave_in_blk = threadIdx.x >> 5;
  const int wave        = blockIdx.x * (blockDim.x >> 5) + wave_in_blk;
  const int n   = lane & 15;   // column index within a 16-wide tile (channel)
  const int hi  = lane >> 4;   // lane half: selects K (A/B) or row+8 (C/D)
  const int batch0 = wave * 16;

  float* hT = lds_h[wave_in_blk];

  // ---------------- per-lane constant weights (registers) ----------------
  // Gate tile t covers channels 16t..16t+15 (i,f,g,o in PyTorch order).
  float biasg[4];               // (b_ih + b_hh) broadcast per gate tile
  float wb0[4], wb1[4];         // W_ih^T as B-matrix (K=4, k=3 zero-padded)
  float whb0[4][4], whb1[4][4]; // W_hh^T as B-matrix, 4 K-chunks per tile
#pragma unroll
  for (int t = 0; t < 4; ++t) {
    const int g = 16 * t + n;
    biasg[t] = b_ih[g] + b_hh[g];
    // B layout: VGPR0 holds k = 2*hi (0 or 2), VGPR1 holds k = 2*hi+1 (1 or 3)
    wb0[t] = W_ih[g * I_ + 2 * hi];                 // k=0 / k=2 (< I_)
    wb1[t] = hi ? 0.0f : W_ih[g * I_ + 1];          // k=1 valid, k=3 -> 0
#pragma unroll
    for (int c = 0; c < 4; ++c) {
      whb0[t][c] = W_hh[g * H_ + 4 * c + 2 * hi];
      whb1[t][c] = W_hh[g * H_ + 4 * c + 2 * hi + 1];
    }
  }
  // Output projection: out = h @ W_lin^T (+ b_lin), N padded 5 -> 16.
  float wlb0[4], wlb1[4];
#pragma unroll
  for (int c = 0; c < 4; ++c) {
    wlb0[c] = (n < O_) ? W_lin[n * H_ + 4 * c + 2 * hi] : 0.0f;
    wlb1[c] = (n < O_) ? W_lin[n * H_ + 4 * c + 2 * hi + 1] : 0.0f;
  }
  const float blin = (n < O_) ? b_lin[n] : 0.0f;

  // ---------------- recurrent state ----------------
  v8f cst;                 // cell state, C-layout: row = r + 8*hi, col = n
#pragma unroll
  for (int r = 0; r < 8; ++r) cst[r] = 0.0f;
  v2f hA[4];               // h in A-layout (4 K-chunks), h0 = 0
#pragma unroll
  for (int c = 0; c < 4; ++c) { hA[c][0] = 0.0f; hA[c][1] = 0.0f; }

  const float* xrow = x + (size_t)(batch0 + n) * S_ * I_;
  const int row_off = 8 * hi;

  // x timestep as A-matrix 16x4 (k=3 zero-padded). Loads are unconditional
  // (addresses valid for every lane); the pad is a cndmask, not a branch.
  v2f xa;
  {
    float a0 = xrow[2 * hi];
    float a1 = xrow[1];
    xa[0] = a0;
    xa[1] = hi ? 0.0f : a1;
  }

  for (int s = 0; s < S_; ++s) {
    // -------- software-pipelined fetch of step s+1 (clamped, branch-free)
    const int sn = (s + 1 < S_) ? (s + 1) : (S_ - 1);
    float nx0 = xrow[sn * I_ + 2 * hi];
    float nx1 = xrow[sn * I_ + 1];
    // Prefetch one more step ahead, folded as an immediate offset onto the
    // same base address as the load above (OOB prefetch is speculative ->
    // silently dropped by hardware).
    __builtin_prefetch(&xrow[sn * I_ + I_], 0, 3);

    // gates = bias + x@W_ih^T + h@W_hh^T, one 5-deep WMMA chain per tile
    v8f acc[4];
#pragma unroll
    for (int t = 0; t < 4; ++t) {
      v8f a;
#pragma unroll
      for (int r = 0; r < 8; ++r) a[r] = biasg[t];
      v2f wb; wb[0] = wb0[t]; wb[1] = wb1[t];
      a = WMMA_F32_16x16x4(xa, wb, a);
#pragma unroll
      for (int c = 0; c < 4; ++c) {
        v2f w; w[0] = whb0[t][c]; w[1] = whb1[t][c];
        a = WMMA_F32_16x16x4(hA[c], w, a);
      }
      acc[t] = a;
    }

    // elementwise LSTM cell (tile0=i, 1=f, 2=g, 3=o), store h into LDS
#pragma unroll
    for (int r = 0; r < 8; ++r) {
      float ig = sigmoid_f(acc[0][r]);
      float fg = sigmoid_f(acc[1][r]);
      float gg = tanh_f(acc[2][r]);
      float og = sigmoid_f(acc[3][r]);
      float cv = fg * cst[r] + ig * gg;
      cst[r] = cv;
      float hv = og * tanh_f(cv);
      hT[(r + row_off) * 16 + n] = hv;   // C-layout -> LDS [row][col]
    }
    __builtin_amdgcn_wave_barrier();     // keep DS stores before DS loads
    // reload h in A-layout for the output matmul and the next timestep
#pragma unroll
    for (int c = 0; c < 4; ++c) {
      hA[c] = *(const v2f*)&hT[n * 16 + 4 * c + 2 * hi];  // ds_load_b64
    }

    // output projection: 4-deep WMMA chain, then clamp and store
    v8f oacc;
#pragma unroll
    for (int r = 0; r < 8; ++r) oacc[r] = blin;
#pragma unroll
    for (int c = 0; c < 4; ++c) {
      v2f w; w[0] = wlb0[c]; w[1] = wlb1[c];
      oacc = WMMA_F32_16x16x4(hA[c], w, oacc);
    }
    if (n < O_) {
      float* orow = out + ((size_t)(batch0 + row_off) * S_ + s) * O_ + n;
#pragma unroll
      for (int r = 0; r < 8; ++r) {
        float v = oacc[r];
        v = fminf(10.0f, fmaxf(-10.0f, v));
        orow[(size_t)r * S_ * O_] = v;
      }
    }

    // rotate the pipelined x operand into place
    xa[0] = nx0;
    xa[1] = hi ? 0.0f : nx1;
  }
}

extern "C" void kernel_launch(void* const* d_in, const int* in_sizes, int n_in,
                              void* d_out, int out_size, void* d_ws, size_t ws_size,
                              hipStream_t stream) {
  const float* x     = (const float*)d_in[0];
  const float* W_ih  = (const float*)d_in[1];
  const float* W_hh  = (const float*)d_in[2];
  const float* b_ih  = (const float*)d_in[3];
  const float* b_hh  = (const float*)d_in[4];
  const float* W_lin = (const float*)d_in[5];
  const float* b_lin = (const float*)d_in[6];
  float* out = (float*)d_out;

  const int Bn = in_sizes[0] / (S_ * I_);   // 4096
  const int waves = Bn / 16;                // 256 waves, 16 rows each
  dim3 block(128);                          // 4 waves per block
  dim3 grid(waves / 4);                     // 64 blocks
  hipLaunchKernelGGL(lstm_wmma_f32_kernel, grid, block, 0, stream,
                     x, W_ih, W_hh, b_ih, b_hh, W_lin, b_lin, out);
}